// EEGGraphAttentionNetwork_22565758173794
// MI455X (gfx1250) — compile-verified
//
#include <hip/hip_runtime.h>
#include <math.h>

#define NN   50000
#define EE   400000
#define ETOT 450000   // EE + NN self loops
#define FF   128
#define HH   8
#define CC   64
#define DD1  512      // HH*CC
#define GG   500
#define NEGS 0.2f
#define EPSB 1e-5f

// GEMM blocking
#define BM 128        // rows per block  (8 waves x 16)
#define BN 64         // cols per block  (4 col tiles per wave)
#define KT 32         // k chunk staged in LDS
#define BPAD 16       // pair-row padding: stride 80 pairs == 32 mod 64 banks

typedef float v2f __attribute__((ext_vector_type(2)));
typedef float v8f __attribute__((ext_vector_type(8)));

static inline int cdiv(long a, long b) { return (int)((a + b - 1) / b); }

// ---------------- float atomic max (sign-split trick) ----------------
__device__ __forceinline__ void atomicMaxF(float* addr, float val) {
  if (val >= 0.0f) atomicMax((int*)addr, __float_as_int(val));
  else             atomicMin((unsigned int*)addr, (unsigned int)__float_as_int(val));
}

// ---------------- fills ----------------
__global__ void fill_kernel(float* __restrict__ p, float v, int n) {
  int i = blockIdx.x * blockDim.x + threadIdx.x;
  if (i < n) p[i] = v;
}

__global__ void bias_fill_kernel(float* __restrict__ out, const float* __restrict__ b,
                                 int D, int total) {
  int i = blockIdx.x * blockDim.x + threadIdx.x;
  if (i < total) out[i] = b[i % D];
}

// ---------------- fp32 WMMA GEMM: D[M x Nc] = A[M x K] * B[K x Nc] ----------------
// row-major; M%16==0, K%KT==0, Nc%BN==0. Block = 8 waves; block tile = 128x64.
// Wave w owns 16 rows, all 64 cols (4 accumulators). B panel staged in LDS in
// K-pair-interleaved layout: Bs[k/2][n][0..1] = B[k..k+1][n], so each WMMA B
// fragment is one contiguous ds_load_b64 into an even VGPR pair.
// A frag (16x4 f32): lane l -> row (l&15), k = 2*(l>>4)+{0,1}; B frag mirrors in K.
// C/D (16x16 f32): vgpr r -> row r + 8*(l>>4), col (l&15).
__global__ __launch_bounds__(256) void gemm_wmma_f32(const float* __restrict__ A,
                                                     const float* __restrict__ B,
                                                     float* __restrict__ D,
                                                     int M, int K, int Nc) {
  __shared__ float Bs[KT / 2][BN + BPAD][2];   // 16 x 80 x 2 floats = 10 KB

  const int wave = threadIdx.x >> 5;
  const int lane = threadIdx.x & 31;
  const int rowTiles = M >> 4;
  const int rowTile = blockIdx.x * 8 + wave;   // 16-row tile owned by this wave
  const int n0 = blockIdx.y * BN;              // first col of block tile
  const bool active = (rowTile < rowTiles);    // wave-uniform

  const int mA  = lane & 15;
  const int kHi = (lane >> 4) << 1;            // 0 or 2
  const int nB  = lane & 15;

  v8f acc[4];
#pragma unroll
  for (int ct = 0; ct < 4; ++ct) acc[ct] = (v8f){0.f,0.f,0.f,0.f,0.f,0.f,0.f,0.f};

  const float* Arow = A + (long)(rowTile * 16 + mA) * K;

  for (int kc = 0; kc < K; kc += KT) {
    __syncthreads();                           // previous chunk fully consumed
    // cooperative, coalesced B panel load: KT x BN = 2048 floats, 8 per thread
    for (int t = threadIdx.x; t < KT * BN; t += 256) {
      int kk = t >> 6;                         // t / BN
      int nn = t & (BN - 1);
      Bs[kk >> 1][nn][kk & 1] = B[(long)(kc + kk) * Nc + n0 + nn];
    }
    __syncthreads();

    if (active) {
#pragma unroll
      for (int kk = 0; kk < KT; kk += 4) {
        v2f a = *(const v2f*)(Arow + kc + kk + kHi);       // 8B global vector load
        const int kp = (kk + kHi) >> 1;                    // pair row (unroll-const + kHi)
#pragma unroll
        for (int ct = 0; ct < 4; ++ct) {
          v2f b = *(const v2f*)&Bs[kp][ct * 16 + nB][0];   // single ds_load_b64
          // 8 args: (neg_a, A, neg_b, B, c_mod, C, reuse_a, reuse_b)
          acc[ct] = __builtin_amdgcn_wmma_f32_16x16x4_f32(false, a, false, b, (short)0,
                                                          acc[ct], false, false);
        }
      }
    }
  }

  if (active) {
    const int rBase = rowTile * 16 + ((lane >> 4) << 3);
    const int cIdx  = lane & 15;
#pragma unroll
    for (int ct = 0; ct < 4; ++ct)
#pragma unroll
      for (int r = 0; r < 8; ++r)
        D[(long)(rBase + r) * Nc + n0 + ct * 16 + cIdx] = acc[ct][r];
  }
}

// ---------------- per-(node,head) attention scores ----------------
__global__ void attn_scores(const float* __restrict__ xp, const float* __restrict__ as_,
                            const float* __restrict__ ad_, float* __restrict__ ssrc,
                            float* __restrict__ sdst, int heads, int ch) {
  int i = blockIdx.x * blockDim.x + threadIdx.x;
  if (i >= NN * heads) return;
  int n = i / heads, h = i - n * heads;
  const float* v = xp + (long)(n * heads + h) * ch;
  float ss = 0.f, sd = 0.f;
  for (int c = 0; c < ch; ++c) {
    float xv = v[c];
    ss += xv * as_[h * ch + c];
    sd += xv * ad_[h * ch + c];
  }
  ssrc[i] = ss;
  sdst[i] = sd;
}

__device__ __forceinline__ void edge_sd(const int* ei, int e, int& s, int& d) {
  if (e < EE) { s = ei[e]; d = ei[EE + e]; }
  else        { s = e - EE; d = e - EE; }          // self loop
}

// ---------------- edge passes: segment max / sum / alpha ----------------
__global__ void edge_max_k(const int* __restrict__ ei, const float* __restrict__ ssrc,
                           const float* __restrict__ sdst, float* __restrict__ amax,
                           int heads) {
  int i = blockIdx.x * blockDim.x + threadIdx.x;
  if (i >= ETOT * heads) return;
  int e = i / heads, h = i - e * heads;
  int s, d; edge_sd(ei, e, s, d);
  float al = ssrc[s * heads + h] + sdst[d * heads + h];
  al = (al > 0.f) ? al : NEGS * al;
  atomicMaxF(&amax[d * heads + h], al);
}

__global__ void edge_sum_k(const int* __restrict__ ei, const float* __restrict__ ssrc,
                           const float* __restrict__ sdst, const float* __restrict__ amax,
                           float* __restrict__ den, int heads) {
  int i = blockIdx.x * blockDim.x + threadIdx.x;
  if (i >= ETOT * heads) return;
  int e = i / heads, h = i - e * heads;
  int s, d; edge_sd(ei, e, s, d);
  float al = ssrc[s * heads + h] + sdst[d * heads + h];
  al = (al > 0.f) ? al : NEGS * al;
  atomicAdd(&den[d * heads + h], expf(al - amax[d * heads + h]));
}

__global__ void edge_alpha_k(const int* __restrict__ ei, const float* __restrict__ ssrc,
                             const float* __restrict__ sdst, const float* __restrict__ amax,
                             const float* __restrict__ den, float* __restrict__ alpha,
                             int heads) {
  int i = blockIdx.x * blockDim.x + threadIdx.x;
  if (i >= ETOT * heads) return;
  int e = i / heads, h = i - e * heads;
  int s, d; edge_sd(ei, e, s, d);
  float al = ssrc[s * heads + h] + sdst[d * heads + h];
  al = (al > 0.f) ? al : NEGS * al;
  float ex = expf(al - amax[d * heads + h]);
  alpha[i] = ex / (den[d * heads + h] + 1e-16f);
}

// ---------------- scatter: hout[dst] += xp[src] * alpha ----------------
__global__ void edge_scatter_k(const int* __restrict__ ei, const float* __restrict__ xp,
                               const float* __restrict__ alpha, float* __restrict__ hout,
                               int heads, int ch, int Dout) {
  long idx = (long)blockIdx.x * blockDim.x + threadIdx.x;
  if (idx >= (long)ETOT * Dout) return;
  int e = (int)(idx / Dout);
  int j = (int)(idx - (long)e * Dout);
  int h = j / ch;
  int s, d; edge_sd(ei, e, s, d);
  float v = xp[(long)s * Dout + j] * alpha[e * heads + h];
  atomicAdd(&hout[(long)d * Dout + j], v);
}

// ---------------- BatchNorm stats (one channel per thread, coalesced rows) ----------------
__global__ void bn_sums_k(const float* __restrict__ h, float* __restrict__ sum,
                          float* __restrict__ sumsq, int D, int rowsPerBlock) {
  int ch = threadIdx.x;                 // blockDim.x == D
  int r0 = blockIdx.x * rowsPerBlock;
  int r1 = r0 + rowsPerBlock; if (r1 > NN) r1 = NN;
  float s = 0.f, q = 0.f;
  for (int r = r0; r < r1; ++r) {
    float x = h[(long)r * D + ch];
    s += x; q += x * x;
  }
  atomicAdd(&sum[ch], s);
  atomicAdd(&sumsq[ch], q);
}

// ---------------- BN + ELU (in place) ----------------
__global__ void bn_elu_k(float* __restrict__ h, const float* __restrict__ sum,
                         const float* __restrict__ sumsq, const float* __restrict__ g,
                         const float* __restrict__ be, int D, long total) {
  long i = (long)blockIdx.x * blockDim.x + threadIdx.x;
  if (i >= total) return;
  int d = (int)(i % D);
  float mu  = sum[d] * (1.0f / NN);
  float var = sumsq[d] * (1.0f / NN) - mu * mu;
  float sc  = g[d] * rsqrtf(var + EPSB);
  float y   = (h[i] - mu) * sc + be[d];
  h[i] = (y > 0.f) ? y : expm1f(y);
}

// ---------------- global mean pool ----------------
__global__ void pool_accum_k(const float* __restrict__ h, const int* __restrict__ batch,
                             float* __restrict__ pooled) {
  int i = blockIdx.x * blockDim.x + threadIdx.x;
  if (i >= NN * CC) return;
  int n = i / CC, c = i - n * CC;
  atomicAdd(&pooled[batch[n] * CC + c], h[i]);
}

__global__ void count_accum_k(const int* __restrict__ batch, float* __restrict__ cnt) {
  int n = blockIdx.x * blockDim.x + threadIdx.x;
  if (n >= NN) return;
  atomicAdd(&cnt[batch[n]], 1.0f);
}

// ---------------- tiny MLP head ----------------
__global__ void mlp_k(const float* __restrict__ pooled, const float* __restrict__ cnt,
                      const float* __restrict__ fw1, const float* __restrict__ fb1,
                      const float* __restrict__ fw2, const float* __restrict__ fb2,
                      float* __restrict__ out) {
  int g = blockIdx.x * blockDim.x + threadIdx.x;
  if (g >= GG) return;
  float inv = 1.0f / fmaxf(cnt[g], 1.0f);
  float hg[CC];
#pragma unroll
  for (int c = 0; c < CC; ++c) hg[c] = pooled[g * CC + c] * inv;
  float o0 = fb2[0], o1 = fb2[1];
  for (int j = 0; j < CC / 2; ++j) {
    float a = fb1[j];
    for (int c = 0; c < CC; ++c) a += hg[c] * fw1[c * (CC / 2) + j];
    a = fmaxf(a, 0.f);
    o0 += a * fw2[j * 2 + 0];
    o1 += a * fw2[j * 2 + 1];
  }
  out[g * 2 + 0] = o0;
  out[g * 2 + 1] = o1;
}

extern "C" void kernel_launch(void* const* d_in, const int* in_sizes, int n_in,
                              void* d_out, int out_size, void* d_ws, size_t ws_size,
                              hipStream_t stream) {
  const float* x     = (const float*)d_in[0];
  const int*   ei    = (const int*)d_in[1];
  const int*   batch = (const int*)d_in[2];
  const float* W1 = (const float*)d_in[3];
  const float* as1 = (const float*)d_in[4];
  const float* ad1 = (const float*)d_in[5];
  const float* b1  = (const float*)d_in[6];
  const float* W2 = (const float*)d_in[7];
  const float* as2 = (const float*)d_in[8];
  const float* ad2 = (const float*)d_in[9];
  const float* b2  = (const float*)d_in[10];
  const float* W3 = (const float*)d_in[11];
  const float* as3 = (const float*)d_in[12];
  const float* ad3 = (const float*)d_in[13];
  const float* b3  = (const float*)d_in[14];
  const float* g1 = (const float*)d_in[15]; const float* be1 = (const float*)d_in[16];
  const float* g2 = (const float*)d_in[17]; const float* be2 = (const float*)d_in[18];
  const float* g3 = (const float*)d_in[19]; const float* be3 = (const float*)d_in[20];
  const float* fw1 = (const float*)d_in[21]; const float* fb1 = (const float*)d_in[22];
  const float* fw2 = (const float*)d_in[23]; const float* fb2 = (const float*)d_in[24];
  float* outp = (float*)d_out;

  // workspace carve-up (floats), ~330 MB total
  float* ws    = (float*)d_ws;
  float* bufA  = ws;                       // N*512 (xp)
  float* bufB  = bufA + (long)NN * DD1;    // N*512
  float* bufC  = bufB + (long)NN * DD1;    // N*512
  float* ssrc  = bufC + (long)NN * DD1;    // N*H
  float* sdst  = ssrc + (long)NN * HH;
  float* amax  = sdst + (long)NN * HH;
  float* den   = amax + (long)NN * HH;
  float* alpha = den  + (long)NN * HH;     // ETOT*H
  float* bnsum = alpha + (long)ETOT * HH;
  float* bnsq  = bnsum + DD1;
  float* pooled= bnsq + DD1;               // G*C
  float* cnt   = pooled + (long)GG * CC;   // G

  auto layer = [&](const float* hin, int Din, const float* W, const float* as_,
                   const float* ad_, const float* b_, const float* g_, const float* be_,
                   int heads, int ch, float* xp, float* hout) {
    const int Dout = heads * ch;
    // 1) xp = hin @ W  (fp32 WMMA, LDS-staged B panel)
    dim3 ggrid(cdiv(NN / 16, 8), Dout / BN);
    gemm_wmma_f32<<<ggrid, 256, 0, stream>>>(hin, W, xp, NN, Din, Dout);
    // 2) attention scores
    attn_scores<<<cdiv((long)NN * heads, 256), 256, 0, stream>>>(xp, as_, ad_, ssrc, sdst,
                                                                 heads, ch);
    // 3) inits
    fill_kernel<<<cdiv((long)NN * heads, 256), 256, 0, stream>>>(amax, -INFINITY, NN * heads);
    fill_kernel<<<cdiv((long)NN * heads, 256), 256, 0, stream>>>(den, 0.f, NN * heads);
    bias_fill_kernel<<<cdiv((long)NN * Dout, 256), 256, 0, stream>>>(hout, b_, Dout,
                                                                     NN * Dout);
    // 4) segment softmax over dst
    const int eb = cdiv((long)ETOT * heads, 256);
    edge_max_k  <<<eb, 256, 0, stream>>>(ei, ssrc, sdst, amax, heads);
    edge_sum_k  <<<eb, 256, 0, stream>>>(ei, ssrc, sdst, amax, den, heads);
    edge_alpha_k<<<eb, 256, 0, stream>>>(ei, ssrc, sdst, amax, den, alpha, heads);
    // 5) weighted scatter
    edge_scatter_k<<<cdiv((long)ETOT * Dout, 256), 256, 0, stream>>>(ei, xp, alpha, hout,
                                                                     heads, ch, Dout);
    // 6) BatchNorm + ELU
    fill_kernel<<<cdiv(Dout, 256), 256, 0, stream>>>(bnsum, 0.f, Dout);
    fill_kernel<<<cdiv(Dout, 256), 256, 0, stream>>>(bnsq, 0.f, Dout);
    bn_sums_k<<<cdiv(NN, 256), Dout, 0, stream>>>(hout, bnsum, bnsq, Dout, 256);
    bn_elu_k<<<cdiv((long)NN * Dout, 256), 256, 0, stream>>>(hout, bnsum, bnsq, g_, be_,
                                                             Dout, (long)NN * Dout);
  };

  layer(x,    FF,  W1, as1, ad1, b1, g1, be1, HH, CC, bufA, bufB);
  layer(bufB, DD1, W2, as2, ad2, b2, g2, be2, HH, CC, bufA, bufC);
  layer(bufC, DD1, W3, as3, ad3, b3, g3, be3, 1,  CC, bufA, bufB);

  // global mean pool + MLP head
  fill_kernel<<<cdiv((long)GG * CC, 256), 256, 0, stream>>>(pooled, 0.f, GG * CC);
  fill_kernel<<<cdiv(GG, 256), 256, 0, stream>>>(cnt, 0.f, GG);
  pool_accum_k<<<cdiv((long)NN * CC, 256), 256, 0, stream>>>(bufB, batch, pooled);
  count_accum_k<<<cdiv(NN, 256), 256, 0, stream>>>(batch, cnt);
  mlp_k<<<cdiv(GG, 64), 64, 0, stream>>>(pooled, cnt, fw1, fb1, fw2, fb2, outp);
}